// hgnn_27444841021697
// MI455X (gfx1250) — compile-verified
//
#include <hip/hip_runtime.h>
#include <hip/hip_bf16.h>

typedef float v2f __attribute__((ext_vector_type(2)));
typedef float v8f __attribute__((ext_vector_type(8)));

// ---------------------------------------------------------------------------
// V_WMMA_F32_16X16X4_F32 chained x4 => C += A(16x16) * B(16x16), all f32.
// Per-lane operand layout (ISA 7.12.2, 32-bit A 16x4):
//   step s: lanes 0-15  (half=0) supply K = 4s+0, 4s+1 for row m = lane
//           lanes 16-31 (half=1) supply K = 4s+2, 4s+3 for row m = lane-16
// The half-dependent element choice is folded into the LOAD ADDRESS (float2
// at column 4s+2*half), not register indexing, so no cndmask select trees.
// C/D: 8 VGPRs, vgpr r = element [m = r + 8*half][n = lane&15].
// ---------------------------------------------------------------------------
__device__ __forceinline__ v8f wmma_chain(const v2f (&av)[4], const v2f (&bv)[4],
                                          v8f c) {
#pragma unroll
  for (int s = 0; s < 4; ++s) {
    c = __builtin_amdgcn_wmma_f32_16x16x4_f32(false, av[s], false, bv[s],
                                              (short)0, c, false, false);
  }
  return c;
}

// Load this lane's 4 A-pairs from a row-major 16-float row, scaled.
__device__ __forceinline__ void load_a_pairs(const float* __restrict__ rowptr,
                                             int half, float scale,
                                             v2f (&av)[4]) {
#pragma unroll
  for (int s = 0; s < 4; ++s) {
    float2 q = *reinterpret_cast<const float2*>(rowptr + 4 * s + 2 * half);
    av[s][0] = q.x * scale;
    av[s][1] = q.y * scale;
  }
}

// Load this lane's 4 B-pairs = column n of a row-major 16x16 weight matrix.
__device__ __forceinline__ void load_b_col(const float* __restrict__ W, int n,
                                           int half, v2f (&bv)[4]) {
#pragma unroll
  for (int s = 0; s < 4; ++s) {
    int k = 4 * s + 2 * half;
    bv[s][0] = W[k * 16 + n];
    bv[s][1] = W[(k + 1) * 16 + n];
  }
}

// ---------------------------------------------------------------------------
__global__ void hgnn_zero_kernel(float* __restrict__ p, long long n) {
  long long i = (long long)blockIdx.x * blockDim.x + threadIdx.x;
  long long stride = (long long)gridDim.x * blockDim.x;
  for (; i < n; i += stride) p[i] = 0.0f;
}

// out[row, c] = feat[row, :F] @ W[F,16] + b[16]
__global__ void hgnn_embed_kernel(const float* __restrict__ feat, int F,
                                  const float* __restrict__ Wm,
                                  const float* __restrict__ b,
                                  float* __restrict__ out, long long nrows) {
  long long tid = (long long)blockIdx.x * blockDim.x + threadIdx.x;
  if (tid >= nrows * 16) return;
  long long row = tid >> 4;
  int c = (int)(tid & 15);
  float acc = b[c];
  for (int f = 0; f < F; ++f) acc += feat[row * F + f] * Wm[f * 16 + c];
  out[tid] = acc;
}

__global__ void hgnn_degree_kernel(const int* __restrict__ idx, long long n,
                                   float* __restrict__ deg) {
  long long i = (long long)blockIdx.x * blockDim.x + threadIdx.x;
  if (i >= n) return;
  atomicAdd(&deg[idx[i]], 1.0f);
}

// d0..d3 -> rsqrt(max(d,1)) ; d4 -> 1/max(d,1)
__global__ void hgnn_finalize_deg_kernel(float* d0, float* d1, float* d2,
                                         float* d3, float* d4, long long n) {
  long long i = (long long)blockIdx.x * blockDim.x + threadIdx.x;
  if (i >= n) return;
  d0[i] = rsqrtf(fmaxf(d0[i], 1.0f));
  d1[i] = rsqrtf(fmaxf(d1[i], 1.0f));
  d2[i] = rsqrtf(fmaxf(d2[i], 1.0f));
  d3[i] = rsqrtf(fmaxf(d3[i], 1.0f));
  d4[i] = 1.0f / fmaxf(d4[i], 1.0f);
}

// One wave per 16-row tile: x_pre = (hj * rs_out_pre)@W_pre,
//                           x_nxt = (hj * rs_out_nxt)@W_nxt
__global__ __launch_bounds__(256) void hgnn_transform_kernel(
    const float* __restrict__ hj, const float* __restrict__ rs_out_pre,
    const float* __restrict__ rs_out_nxt, const float* __restrict__ Wp,
    const float* __restrict__ Wn, float* __restrict__ x_pre,
    float* __restrict__ x_nxt) {
  int gw = (blockIdx.x * 256 + threadIdx.x) >> 5;  // tile index
  int lane = threadIdx.x & 31;
  int half = lane >> 4;
  int mn = lane & 15;
  long long base = (long long)gw * 16;
  long long row = base + mn;

  float sp = rs_out_pre[row], sn = rs_out_nxt[row];
  const float* rowptr = hj + row * 16;
  v2f ap[4], an[4], bp[4], bn[4];
  load_a_pairs(rowptr, half, sp, ap);
  load_a_pairs(rowptr, half, sn, an);
  load_b_col(Wp, mn, half, bp);
  load_b_col(Wn, mn, half, bn);

  v8f cz = {0.f, 0.f, 0.f, 0.f, 0.f, 0.f, 0.f, 0.f};
  v8f cp = wmma_chain(ap, bp, cz);
  v8f cn = wmma_chain(an, bn, cz);
#pragma unroll
  for (int r = 0; r < 8; ++r) {
    long long rr = base + r + 8 * half;
    x_pre[rr * 16 + mn] = cp[r];
    x_nxt[rr * 16 + mn] = cn[r];
  }
}

// Edge-parallel segment-sum scatter: agg[dst[e], c] += x[src[e], c]
__global__ void hgnn_scatter_kernel(const float* __restrict__ x,
                                    const int* __restrict__ src,
                                    const int* __restrict__ dst,
                                    float* __restrict__ agg, long long nE) {
  long long tid = (long long)blockIdx.x * blockDim.x + threadIdx.x;
  long long e = tid >> 4;
  if (e >= nE) return;
  int c = (int)(tid & 15);
  if (e + 2048 < nE) {
    __builtin_prefetch(&src[e + 2048], 0, 1);
    __builtin_prefetch(&dst[e + 2048], 0, 1);
  }
  long long s = src[e], d = dst[e];
  atomicAdd(&agg[d * 16 + c], x[s * 16 + c]);
}

// One wave per tile:
// hj_new = agg_pre*rs_in_pre + b_pre + agg_nxt*rs_in_nxt + b_nxt
//          + hj@W_self + (agg_proc*inv_deg_proc)@W_neigh + b_sage
__global__ __launch_bounds__(256) void hgnn_combine_kernel(
    const float* __restrict__ hj, const float* __restrict__ agg_pre,
    const float* __restrict__ agg_nxt, const float* __restrict__ agg_proc,
    const float* __restrict__ rs_in_pre, const float* __restrict__ rs_in_nxt,
    const float* __restrict__ inv_proc, const float* __restrict__ Wself,
    const float* __restrict__ Wneigh, const float* __restrict__ b_pre,
    const float* __restrict__ b_nxt, const float* __restrict__ b_sage,
    float* __restrict__ out) {
  int gw = (blockIdx.x * 256 + threadIdx.x) >> 5;
  int lane = threadIdx.x & 31;
  int half = lane >> 4;
  int mn = lane & 15;
  long long base = (long long)gw * 16;

  float bias = b_pre[mn] + b_nxt[mn] + b_sage[mn];
  v8f c;
#pragma unroll
  for (int r = 0; r < 8; ++r) {
    long long rr = base + r + 8 * half;
    c[r] = agg_pre[rr * 16 + mn] * rs_in_pre[rr] +
           agg_nxt[rr * 16 + mn] * rs_in_nxt[rr] + bias;
  }

  long long row = base + mn;
  v2f a1[4], a2[4], bs[4], bn2[4];
  load_a_pairs(hj + row * 16, half, 1.0f, a1);
  load_a_pairs(agg_proc + row * 16, half, inv_proc[row], a2);
  load_b_col(Wself, mn, half, bs);
  load_b_col(Wneigh, mn, half, bn2);

  c = wmma_chain(a1, bs, c);
  c = wmma_chain(a2, bn2, c);
#pragma unroll
  for (int r = 0; r < 8; ++r) {
    long long rr = base + r + 8 * half;
    out[rr * 16 + mn] = c[r];
  }
}

// One wave per (subgraph, job-tile): scores = Hj_tile[16,16] @ Hw[16,16]^T
__global__ __launch_bounds__(256) void hgnn_readout_kernel(
    const float* __restrict__ hj, const float* __restrict__ hw,
    float* __restrict__ out, int jobs_per_b) {
  int gw = (blockIdx.x * 256 + threadIdx.x) >> 5;
  int tiles_per_b = jobs_per_b >> 4;
  int b = gw / tiles_per_b;
  int jt = gw - b * tiles_per_b;
  int lane = threadIdx.x & 31;
  int half = lane >> 4;
  int mn = lane & 15;

  long long jrow = (long long)b * jobs_per_b + jt * 16 + mn;
  long long wrow = (long long)b * 16 + mn;  // B[k][n] = hw[b*16+n][k]
  v2f a[4], bb[4];
  load_a_pairs(hj + jrow * 16, half, 1.0f, a);
  load_a_pairs(hw + wrow * 16, half, 1.0f, bb);  // same pair pattern as A

  v8f c = {0.f, 0.f, 0.f, 0.f, 0.f, 0.f, 0.f, 0.f};
  c = wmma_chain(a, bb, c);
#pragma unroll
  for (int r = 0; r < 8; ++r) {
    long long j = (long long)jt * 16 + r + 8 * half;
    out[((long long)b * jobs_per_b + j) * 16 + mn] = c[r];
  }
}

// ---------------------------------------------------------------------------
extern "C" void kernel_launch(void* const* d_in, const int* in_sizes, int n_in,
                              void* d_out, int out_size, void* d_ws,
                              size_t ws_size, hipStream_t stream) {
  (void)n_in; (void)out_size; (void)ws_size;
  const float* job_feat     = (const float*)d_in[0];
  const float* worker_feat  = (const float*)d_in[1];
  const int*   pre_src      = (const int*)d_in[2];
  const int*   pre_dst      = (const int*)d_in[3];
  const int*   nxt_src      = (const int*)d_in[4];
  const int*   nxt_dst      = (const int*)d_in[5];
  const int*   proc_src     = (const int*)d_in[6];
  const int*   proc_dst     = (const int*)d_in[7];
  const float* W_emb_job    = (const float*)d_in[8];
  const float* b_emb_job    = (const float*)d_in[9];
  const float* W_emb_worker = (const float*)d_in[10];
  const float* b_emb_worker = (const float*)d_in[11];
  const float* W_pre        = (const float*)d_in[12];
  const float* b_pre        = (const float*)d_in[13];
  const float* W_nxt        = (const float*)d_in[14];
  const float* b_nxt        = (const float*)d_in[15];
  const float* W_self       = (const float*)d_in[16];
  const float* W_neigh      = (const float*)d_in[17];
  const float* b_sage       = (const float*)d_in[18];

  const long long NJ = in_sizes[0] / 7;   // 131072
  const long long NW = in_sizes[1] / 3;   // 4096
  const long long E_pre  = in_sizes[2];
  const long long E_nxt  = in_sizes[4];
  const long long E_proc = in_sizes[6];
  const int jobs_per_b = (int)(NJ / (NW / 16));  // 512

  // -------- workspace layout (floats) --------
  float* ws = (float*)d_ws;
  float* hjA      = ws; ws += NJ * 16;
  float* hjB      = ws; ws += NJ * 16;
  float* hw       = ws; ws += NW * 16;
  float* x_pre    = ws; ws += NJ * 16;
  float* x_nxt    = ws; ws += NJ * 16;
  float* agg_pre  = ws; ws += NJ * 16;   // agg_* contiguous: one zero pass
  float* agg_nxt  = ws; ws += NJ * 16;
  float* agg_proc = ws; ws += NJ * 16;
  float* rs_out_pre = ws; ws += NJ;      // degree buffers contiguous
  float* rs_in_pre  = ws; ws += NJ;
  float* rs_out_nxt = ws; ws += NJ;
  float* rs_in_nxt  = ws; ws += NJ;
  float* inv_proc   = ws; ws += NJ;

  // -------- degrees (loop-invariant) --------
  hgnn_zero_kernel<<<2048, 256, 0, stream>>>(rs_out_pre, 5 * NJ);
  {
    unsigned g = (unsigned)((E_pre + 255) / 256);
    hgnn_degree_kernel<<<g, 256, 0, stream>>>(pre_src, E_pre, rs_out_pre);
    hgnn_degree_kernel<<<g, 256, 0, stream>>>(pre_dst, E_pre, rs_in_pre);
    hgnn_degree_kernel<<<g, 256, 0, stream>>>(nxt_src, E_nxt, rs_out_nxt);
    hgnn_degree_kernel<<<g, 256, 0, stream>>>(nxt_dst, E_nxt, rs_in_nxt);
    unsigned gp = (unsigned)((E_proc + 255) / 256);
    hgnn_degree_kernel<<<gp, 256, 0, stream>>>(proc_dst, E_proc, inv_proc);
  }
  hgnn_finalize_deg_kernel<<<(unsigned)((NJ + 255) / 256), 256, 0, stream>>>(
      rs_out_pre, rs_in_pre, rs_out_nxt, rs_in_nxt, inv_proc, NJ);

  // -------- embeddings --------
  hgnn_embed_kernel<<<(unsigned)((NJ * 16 + 255) / 256), 256, 0, stream>>>(
      job_feat, 7, W_emb_job, b_emb_job, hjA, NJ);
  hgnn_embed_kernel<<<(unsigned)((NW * 16 + 255) / 256), 256, 0, stream>>>(
      worker_feat, 3, W_emb_worker, b_emb_worker, hw, NW);

  // -------- message-passing loops --------
  const unsigned tile_blocks = (unsigned)(NJ / 16 / 8);  // 8 waves per block
  float* hj_cur = hjA;
  float* hj_nxt = hjB;
  for (int t = 0; t < 2; ++t) {
    hgnn_transform_kernel<<<tile_blocks, 256, 0, stream>>>(
        hj_cur, rs_out_pre, rs_out_nxt, W_pre, W_nxt, x_pre, x_nxt);
    hgnn_zero_kernel<<<4096, 256, 0, stream>>>(agg_pre, 3 * NJ * 16);
    hgnn_scatter_kernel<<<(unsigned)((E_pre * 16 + 255) / 256), 256, 0, stream>>>(
        x_pre, pre_src, pre_dst, agg_pre, E_pre);
    hgnn_scatter_kernel<<<(unsigned)((E_nxt * 16 + 255) / 256), 256, 0, stream>>>(
        x_nxt, nxt_src, nxt_dst, agg_nxt, E_nxt);
    hgnn_scatter_kernel<<<(unsigned)((E_proc * 16 + 255) / 256), 256, 0, stream>>>(
        hw, proc_src, proc_dst, agg_proc, E_proc);
    hgnn_combine_kernel<<<tile_blocks, 256, 0, stream>>>(
        hj_cur, agg_pre, agg_nxt, agg_proc, rs_in_pre, rs_in_nxt, inv_proc,
        W_self, W_neigh, b_pre, b_nxt, b_sage, hj_nxt);
    float* tmp = hj_cur; hj_cur = hj_nxt; hj_nxt = tmp;
  }

  // -------- readout --------
  hgnn_readout_kernel<<<tile_blocks, 256, 0, stream>>>(
      hj_cur, hw, (float*)d_out, jobs_per_b);
}